// SphericalCheb_7687991460104
// MI455X (gfx1250) — compile-verified
//
#include <hip/hip_runtime.h>

// SphericalChebConv: T0=x, T1=Lx, Tk=2LT(k-1)-T(k-2); out = leaky(sum_k Tk @ Wk + b)
// B=2, V=49152, Fin=Fout=64, K=4, E=8V.

typedef __attribute__((ext_vector_type(2))) float v2f;
typedef __attribute__((ext_vector_type(8))) float v8f;

#define NEG_SLOPE 0.2f

constexpr int Bc   = 2;
constexpr int Vc   = 49152;
constexpr int Fc   = 64;          // Fin == Fout
constexpr int Kc   = 4;
constexpr int Ec   = 8 * Vc;      // 393216 edges
constexpr int Mrows = Bc * Vc;    // 98304 GEMM rows
constexpr int NTOT  = Mrows * Fc; // elements per T buffer / output

// ---------------- init: dst = (src ? -src : 0) ------------------------------
__global__ __launch_bounds__(256) void cheb_init(float* __restrict__ dst,
                                                 const float* __restrict__ src,
                                                 int n) {
  int i = blockIdx.x * 256 + threadIdx.x;
  if (i < n) dst[i] = src ? -src[i] : 0.0f;
}

// ---------------- SpMM scatter: dst += scale * vals * src[cols] -------------
// One wave per (edge, batch); lane handles features {lane, lane+32}.
__global__ __launch_bounds__(256) void cheb_spmm(const int* __restrict__ rows,
                                                 const int* __restrict__ cols,
                                                 const float* __restrict__ vals,
                                                 const float* __restrict__ src,
                                                 float* __restrict__ dst,
                                                 float scale) {
  unsigned gid  = blockIdx.x * 256u + threadIdx.x;
  unsigned wid  = gid >> 5;
  unsigned lane = gid & 31u;
  if (wid >= (unsigned)(Ec * Bc)) return;
  unsigned e = wid % (unsigned)Ec;
  unsigned b = wid / (unsigned)Ec;
  int   r = rows[e];
  int   c = cols[e];
  float v = vals[e] * scale;
  const float* sp = src + ((size_t)b * Vc + c) * Fc + lane;
  float*       dp = dst + ((size_t)b * Vc + r) * Fc + lane;
  atomicAdd(dp,      v * sp[0]);
  atomicAdd(dp + 32, v * sp[32]);
}

// ---------------- fused GEMM + bias + LeakyReLU -----------------------------
// out[m, o] = leaky( sum_c sum_i Tc[m, i] * W[c, i, o] + bias[o] )
// 128 threads = 4 waves; each wave computes a 16(M) x 64(N) strip via
// V_WMMA_F32_16X16X4_F32 (exact fp32). W staged once into LDS, transposed
// into (chunk, K-pair, o, parity) so B-fragment ds_load_b64 is lane-contiguous.
__global__ __launch_bounds__(128) void cheb_gemm(const float* __restrict__ t0,
                                                 const float* __restrict__ t1,
                                                 const float* __restrict__ t2,
                                                 const float* __restrict__ t3,
                                                 const float* __restrict__ w,
                                                 const float* __restrict__ bias,
                                                 float* __restrict__ out) {
  // wl[((c*32 + (K>>1))*64 + o)*2 + (K&1)]  -- 16384 floats = 64 KB LDS
  __shared__ float wl[Kc * 32 * Fc * 2];
  for (int idx = threadIdx.x; idx < Kc * Fc * Fc; idx += 128) {
    int c = idx >> 12;
    int i = (idx >> 6) & 63;
    int o = idx & 63;
    wl[(((c * 32 + (i >> 1)) * 64) + o) * 2 + (i & 1)] = w[idx];
  }
  __syncthreads();

  const int wave   = threadIdx.x >> 5;
  const int lane   = threadIdx.x & 31;
  const int lane16 = lane & 15;
  const int khalf  = lane >> 4;                // 0: K={kk,kk+1}, 1: K={kk+2,kk+3}
  const int m0     = (blockIdx.x * 4 + wave) * 16;

  v8f acc0 = {}, acc1 = {}, acc2 = {}, acc3 = {};
  const float* tb[4] = {t0, t1, t2, t3};

  for (int c = 0; c < 4; ++c) {
    // A fragment: lane reads float2 {K=kk+2*khalf, +1} of row m0+lane16
    const float* Ap = tb[c] + (size_t)(m0 + lane16) * Fc + 2 * khalf;
    // B fragment base in LDS for o = lane16 (+16 per N tile)
    const float* Bp = &wl[((c * 32 + khalf) * 64 + lane16) * 2];
#pragma unroll
    for (int kk = 0; kk < 64; kk += 4) {
      v2f a  = *(const v2f*)(Ap + kk);
      v2f b0 = *(const v2f*)(Bp + kk * 64);
      v2f b1 = *(const v2f*)(Bp + kk * 64 + 32);
      v2f b2 = *(const v2f*)(Bp + kk * 64 + 64);
      v2f b3 = *(const v2f*)(Bp + kk * 64 + 96);
      acc0 = __builtin_amdgcn_wmma_f32_16x16x4_f32(false, a, false, b0, (short)0, acc0, false, false);
      acc1 = __builtin_amdgcn_wmma_f32_16x16x4_f32(false, a, false, b1, (short)0, acc1, false, false);
      acc2 = __builtin_amdgcn_wmma_f32_16x16x4_f32(false, a, false, b2, (short)0, acc2, false, false);
      acc3 = __builtin_amdgcn_wmma_f32_16x16x4_f32(false, a, false, b3, (short)0, acc3, false, false);
    }
  }

  // D layout: VGPR r, lanes 0-15 -> M=r, lanes 16-31 -> M=r+8; N = lane16 (+16n)
#define EPILOGUE(ACC, N)                                                        \
  {                                                                             \
    float bb = bias[(N) * 16 + lane16];                                         \
    _Pragma("unroll") for (int r = 0; r < 8; ++r) {                             \
      float vv = ACC[r] + bb;                                                   \
      vv = vv >= 0.0f ? vv : NEG_SLOPE * vv;                                    \
      out[(size_t)(m0 + r + 8 * khalf) * Fc + (N) * 16 + lane16] = vv;          \
    }                                                                           \
  }
  EPILOGUE(acc0, 0)
  EPILOGUE(acc1, 1)
  EPILOGUE(acc2, 2)
  EPILOGUE(acc3, 3)
#undef EPILOGUE
}

// ---------------------------------------------------------------------------
extern "C" void kernel_launch(void* const* d_in, const int* in_sizes, int n_in,
                              void* d_out, int out_size, void* d_ws, size_t ws_size,
                              hipStream_t stream) {
  const float* x    = (const float*)d_in[0];
  const int*   rows = (const int*)d_in[1];
  const int*   cols = (const int*)d_in[2];
  const float* vals = (const float*)d_in[3];
  const float* w    = (const float*)d_in[4];
  const float* bias = (const float*)d_in[5];
  float*       out  = (float*)d_out;

  float* t1 = (float*)d_ws;          // each T buffer: NTOT floats (~25 MB)
  float* t2 = t1 + NTOT;
  float* t3 = t2 + NTOT;

  const int init_blk = (NTOT + 255) / 256;
  const int spmm_blk = (Ec * Bc) / 8;  // 8 waves (edge,batch) per 256-thread block

  // T1 = L x
  cheb_init<<<init_blk, 256, 0, stream>>>(t1, nullptr, NTOT);
  cheb_spmm<<<spmm_blk, 256, 0, stream>>>(rows, cols, vals, x, t1, 1.0f);
  // T2 = 2 L T1 - x
  cheb_init<<<init_blk, 256, 0, stream>>>(t2, x, NTOT);
  cheb_spmm<<<spmm_blk, 256, 0, stream>>>(rows, cols, vals, t1, t2, 2.0f);
  // T3 = 2 L T2 - T1
  cheb_init<<<init_blk, 256, 0, stream>>>(t3, t1, NTOT);
  cheb_spmm<<<spmm_blk, 256, 0, stream>>>(rows, cols, vals, t2, t3, 2.0f);
  // out = leaky( [x|T1|T2|T3] @ W + bias )
  cheb_gemm<<<Mrows / 64, 128, 0, stream>>>(x, t1, t2, t3, w, bias, out);
}